// GCN_58643483460244
// MI455X (gfx1250) — compile-verified
//
#include <hip/hip_runtime.h>
#include <hip/hip_bf16.h>

// ---------------------------------------------------------------------------
// GCN forward on MI455X (gfx1250, wave32).
//   - dense GEMMs via V_WMMA_F32_16X16X4_F32 (f32 fidelity, CDNA5 matrix path)
//   - edge scatter via native global_atomic_add_f32 (L2-resident: hw/agg are
//     51.2 MB each, well under the 192 MB L2)
// ---------------------------------------------------------------------------

#define N_NODES 100000
#define N_EDGES 1600000
#define F_DIM   64
#define H_DIM   128
#define G_DIM   512
#define N_STEPS 3

typedef float v2f __attribute__((ext_vector_type(2)));
typedef float v8f __attribute__((ext_vector_type(8)));

// ---------------- degree / normalization ----------------------------------

__global__ void init_deg_kernel(float* __restrict__ deg, int n) {
    int i = blockIdx.x * blockDim.x + threadIdx.x;
    if (i < n) deg[i] = 1.0f;               // self-loop
}

__global__ void edge_deg_kernel(const int* __restrict__ dst,
                                float* __restrict__ deg, int e) {
    int i = blockIdx.x * blockDim.x + threadIdx.x;
    if (i < e) atomicAdd(&deg[dst[i]], 1.0f);
}

__global__ void rsqrt_kernel(float* __restrict__ deg, int n) {
    int i = blockIdx.x * blockDim.x + threadIdx.x;
    if (i < n) deg[i] = rsqrtf(deg[i]);     // deg buffer becomes dinv
}

// ---------------- WMMA f32 GEMM: C[M,128] = A[M,K] @ B[K,128] (+bias) -------
//
// 256 threads = 8 wave32s per block. Wave w owns column tile [16w,16w+16).
// B fragments for the whole K loop are preloaded into registers and reused
// across MT row tiles.  VGPR layouts per CDNA5 ISA 7.12.2:
//   A 16x4  : lane<16 -> K=2*half+{0,1} at row=lane&15 (half = lane>>4)
//   B 4x16  : lane<16 -> N=lane&15, K=2*half+{0,1}
//   C 16x16 : vgpr r   -> row = r + 8*half, col = (lane&15) + 16*wave

template <int K, int MT>
__global__ __launch_bounds__(256)
void gemm_wmma_f32_kernel(const float* __restrict__ A,
                          const float* __restrict__ B,
                          const float* __restrict__ bias,
                          float* __restrict__ C,
                          int M, int hasBias) {
    constexpr int KC = K / 4;
    const int lane = threadIdx.x & 31;
    const int wave = threadIdx.x >> 5;       // 0..7 -> column tile
    const int half = lane >> 4;              // 0/1
    const int lidx = lane & 15;
    const int col  = wave * 16 + lidx;

    // Preload B fragments (weights) for this column tile.
    v2f bfrag[KC];
#pragma unroll
    for (int kc = 0; kc < KC; ++kc) {
        const int k0 = kc * 4 + 2 * half;
        v2f b;
        b[0] = B[(size_t)(k0 + 0) * H_DIM + col];
        b[1] = B[(size_t)(k0 + 1) * H_DIM + col];
        bfrag[kc] = b;
    }

    const float badd = hasBias ? bias[col] : 0.0f;
    const int rowBase = blockIdx.x * (16 * MT);

#pragma unroll
    for (int mt = 0; mt < MT; ++mt) {
        const int row0 = rowBase + mt * 16;
        if (row0 >= M) break;                // M is a multiple of 16

        v8f acc = {};
        const float* Ap = A + (size_t)(row0 + lidx) * K + 2 * half;
#pragma unroll
        for (int kc = 0; kc < KC; ++kc) {
            v2f a = *(const v2f*)(Ap + kc * 4);   // 8B-aligned pair
            acc = __builtin_amdgcn_wmma_f32_16x16x4_f32(
                false, a, false, bfrag[kc], (short)0, acc, false, false);
        }

#pragma unroll
        for (int r = 0; r < 8; ++r) {
            const int row = row0 + r + 8 * half;
            C[(size_t)row * H_DIM + col] = acc[r] + badd;
        }
    }
}

// ---------------- message passing ------------------------------------------

// agg = dinv[v]^2 * hw   (self-loop term; also initializes agg, no memset)
__global__ void agg_init_kernel(const float* __restrict__ hw,
                                const float* __restrict__ dinv,
                                float* __restrict__ agg, int n) {
    int i = blockIdx.x * blockDim.x + threadIdx.x;
    if (i < n) {
        const int v = i >> 7;                // / H_DIM
        const float d = dinv[v];
        agg[i] = d * d * hw[i];
    }
}

// One wave per edge; lane l handles features [4l, 4l+4).  hw/agg stay in L2.
__global__ __launch_bounds__(256)
void scatter_kernel(const int* __restrict__ src, const int* __restrict__ dst,
                    const float* __restrict__ dinv,
                    const float* __restrict__ hw, float* __restrict__ agg,
                    int e) {
    const int wave = threadIdx.x >> 5;
    const int lane = threadIdx.x & 31;
    const int ed = blockIdx.x * 8 + wave;
    if (ed >= e) return;
    const int s = src[ed];
    const int d = dst[ed];
    const float w = dinv[s] * dinv[d];
    const float4 v = ((const float4*)(hw + (size_t)s * H_DIM))[lane];
    float* ap = agg + (size_t)d * H_DIM + lane * 4;
    atomicAdd(ap + 0, w * v.x);
    atomicAdd(ap + 1, w * v.y);
    atomicAdd(ap + 2, w * v.z);
    atomicAdd(ap + 3, w * v.w);
}

__global__ void bias_relu_kernel(float* __restrict__ h,
                                 const float* __restrict__ b, int n) {
    int i = blockIdx.x * blockDim.x + threadIdx.x;
    if (i < n) h[i] = fmaxf(h[i] + b[i & (H_DIM - 1)], 0.0f);
}

// ---------------- pooling + readout MLP ------------------------------------

__global__ void zero_kernel(float* __restrict__ p, int n) {
    int i = blockIdx.x * blockDim.x + threadIdx.x;
    if (i < n) p[i] = 0.0f;
}

// block = node, thread = feature
__global__ __launch_bounds__(H_DIM)
void pool_kernel(const float* __restrict__ h, const int* __restrict__ batch,
                 float* __restrict__ gsum, float* __restrict__ cnt) {
    const int n = blockIdx.x;
    const int f = threadIdx.x;
    const int b = batch[n];
    atomicAdd(&gsum[(size_t)b * H_DIM + f], h[(size_t)n * H_DIM + f]);
    if (f == 0) atomicAdd(&cnt[b], 1.0f);
}

// block = graph (512), thread = hidden unit (128)
__global__ __launch_bounds__(H_DIM)
void mlp_kernel(const float* __restrict__ gsum, const float* __restrict__ cnt,
                const float* __restrict__ W1, const float* __restrict__ b1,
                const float* __restrict__ W2, const float* __restrict__ b2,
                float* __restrict__ out) {
    __shared__ float pooled[2 * H_DIM];
    __shared__ float red[H_DIM];
    const int g = blockIdx.x;
    const int f = threadIdx.x;

    const float s = gsum[(size_t)g * H_DIM + f];
    const float c = fmaxf(cnt[g], 1.0f);
    pooled[f] = s;                 // gsum part
    pooled[H_DIM + f] = s / c;     // gmean part
    __syncthreads();

    float acc = b1[f];
#pragma unroll 8
    for (int k = 0; k < 2 * H_DIM; ++k)
        acc = fmaf(pooled[k], W1[(size_t)k * H_DIM + f], acc);

    red[f] = fmaxf(acc, 0.0f) * W2[f];
    __syncthreads();
    for (int st = H_DIM / 2; st > 0; st >>= 1) {
        if (f < st) red[f] += red[f + st];
        __syncthreads();
    }
    if (f == 0) out[g] = red[0] + b2[0];
}

// ---------------- launch ----------------------------------------------------

extern "C" void kernel_launch(void* const* d_in, const int* in_sizes, int n_in,
                              void* d_out, int out_size, void* d_ws, size_t ws_size,
                              hipStream_t stream) {
    const float* x       = (const float*)d_in[0];
    const int*   ei      = (const int*)  d_in[1];   // [2,E] flat
    const int*   batch   = (const int*)  d_in[3];
    const float* W_embed = (const float*)d_in[4];
    const float* b_embed = (const float*)d_in[5];
    const float* Ws      = (const float*)d_in[6];   // [3,H,H]
    const float* bs      = (const float*)d_in[7];   // [3,H]
    const float* W1      = (const float*)d_in[8];
    const float* b1      = (const float*)d_in[9];
    const float* W2      = (const float*)d_in[10];
    const float* b2      = (const float*)d_in[11];
    float* out = (float*)d_out;

    // workspace layout (floats): dinv | h | hw | gsum | cnt
    float* ws   = (float*)d_ws;
    float* dinv = ws;
    float* h    = dinv + ((N_NODES + 255) & ~255);
    float* hw   = h  + (size_t)N_NODES * H_DIM;
    float* gsum = hw + (size_t)N_NODES * H_DIM;
    float* cnt  = gsum + (size_t)G_DIM * H_DIM;     // contiguous after gsum

    const int* src = ei;
    const int* dst = ei + N_EDGES;

    // 1. degrees -> dinv
    init_deg_kernel<<<(N_NODES + 255) / 256, 256, 0, stream>>>(dinv, N_NODES);
    edge_deg_kernel<<<(N_EDGES + 255) / 256, 256, 0, stream>>>(dst, dinv, N_EDGES);
    rsqrt_kernel<<<(N_NODES + 255) / 256, 256, 0, stream>>>(dinv, N_NODES);

    // 2. embed: h = x @ W_embed + b_embed
    constexpr int MT = 4;
    const int gemmGrid = (N_NODES + 16 * MT - 1) / (16 * MT);
    gemm_wmma_f32_kernel<F_DIM, MT><<<gemmGrid, 256, 0, stream>>>(
        x, W_embed, b_embed, h, N_NODES, 1);

    // 3. message-passing steps
    const int nh = N_NODES * H_DIM;
    for (int st = 0; st < N_STEPS; ++st) {
        gemm_wmma_f32_kernel<H_DIM, MT><<<gemmGrid, 256, 0, stream>>>(
            h, Ws + (size_t)st * H_DIM * H_DIM, b_embed, hw, N_NODES, 0);
        agg_init_kernel<<<(nh + 255) / 256, 256, 0, stream>>>(hw, dinv, h, nh);
        scatter_kernel<<<(N_EDGES + 7) / 8, 256, 0, stream>>>(
            src, dst, dinv, hw, h, N_EDGES);
        bias_relu_kernel<<<(nh + 255) / 256, 256, 0, stream>>>(
            h, bs + (size_t)st * H_DIM, nh);
    }

    // 4. pooling (gsum & cnt are contiguous -> single zero pass)
    zero_kernel<<<(G_DIM * H_DIM + G_DIM + 255) / 256, 256, 0, stream>>>(
        gsum, G_DIM * H_DIM + G_DIM);
    pool_kernel<<<N_NODES, H_DIM, 0, stream>>>(h, batch, gsum, cnt);

    // 5. readout MLP
    mlp_kernel<<<G_DIM, H_DIM, 0, stream>>>(gsum, cnt, W1, b1, W2, b2, out);
}